// GCNPre_9448928051676
// MI455X (gfx1250) — compile-verified
//
#include <hip/hip_runtime.h>

typedef float v2f __attribute__((ext_vector_type(2)));
typedef float v8f __attribute__((ext_vector_type(8)));

// ---------------------------------------------------------------------------
// Degree histogram: deg[col[e]] += 1.0f
// ---------------------------------------------------------------------------
__global__ void deg_kernel(const int* __restrict__ col, float* __restrict__ deg, long E) {
    long e = (long)blockIdx.x * blockDim.x + threadIdx.x;
    if (e >= E) return;
    atomicAdd(&deg[col[e]], 1.0f);
}

// dis[i] = rsqrt(deg[i] + 1)   (self-loop included in degree)
__global__ void dis_kernel(const float* __restrict__ deg, float* __restrict__ dis, int n) {
    int i = blockIdx.x * blockDim.x + threadIdx.x;
    if (i >= n) return;
    dis[i] = rsqrtf(deg[i] + 1.0f);
}

// Pack a [K, Nsrc] matrix (K even) into pair-interleaved, zero-padded form:
//   dst[(k/2) * 2*Npad + n*2 + (k&1)] = src[k*Nsrc + n]   (0 for n >= Nsrc)
// so {B[k][n], B[k+1][n]} are contiguous 8 bytes -> one ds_load_b64 fragment.
__global__ void pack_pairs_kernel(const float* __restrict__ src, float* __restrict__ dst,
                                  int K, int Nsrc, int Npad) {
    int idx = blockIdx.x * blockDim.x + threadIdx.x;
    if (idx >= K * Npad) return;
    const int p   = idx / (2 * Npad);
    const int rem = idx % (2 * Npad);
    const int n   = rem >> 1;
    const int r   = rem & 1;
    const int k   = 2 * p + r;
    dst[idx] = (n < Nsrc) ? src[k * Nsrc + n] : 0.0f;
}

// ---------------------------------------------------------------------------
// FP32 WMMA GEMM:  C[M,Nst] = act(A[M,K] @ B[K,*] + bias)
// B is the pair-interleaved packed weight (row stride 2*Nld, Nld % BN == 0).
//
// Block: 256 threads = 8 waves. blockIdx.x -> group of 8 M-tiles (one per
// wave), blockIdx.y -> N-group of BN = 16*NT columns (shared by all waves).
//
// B streams through a double-buffered LDS tile via CDNA5 async global->LDS
// B128 copies (ASYNCcnt); the next chunk's copy overlaps the current chunk's
// WMMA stream; one barrier + one s_wait_asynccnt per K-chunk. The inner loop
// is branch-free: per (kk, nt) it is exactly one ds_load_b64 + one
// v_wmma_f32_16x16x4_f32 (A: one global_load_b64 per kk, reused across NT).
//
// LDS pair-row stride LDP ≡ 32 (mod 64 banks): the two 16-lane halves of a
// wave (reading pair rows p and p+1) hit disjoint 32-bank ranges.
//
// V_WMMA_F32_16X16X4_F32 fragment layout (ISA 7.12.2, 32-bit matrices):
//   A: lane<16 -> row M=lane, {K,K+1}; lane>=16 -> row lane-16, {K+2,K+3}
//   B: lane<16 -> col N=lane, rows {K,K+1}; lane>=16 -> col lane-16, {K+2,K+3}
//   C/D: vgpr v: lane<16 -> (m0+v, N=lane); lane>=16 -> (m0+v+8, N=lane-16)
// ---------------------------------------------------------------------------
template <int NT>
__global__ void __launch_bounds__(256)
wmma_gemm_f32(const float* __restrict__ A, const float* __restrict__ B,
              const float* __restrict__ bias, float* __restrict__ C,
              int M, int K, int Nld, int Nst, int relu) {
    constexpr int BN   = 16 * NT;                          // columns per block
    constexpr int KC   = 32;                               // K chunk in LDS
    constexpr int PR   = KC / 2;                           // pair rows per chunk
    constexpr int LDP  = ((2 * BN + 31) / 64) * 64 + 32;   // 160 (BN=64) / 96 (BN=48)
    constexpr int BUFS = PR * LDP;                         // floats per buffer
    constexpr int CH   = PR * (BN / 2);                    // b128 chunks per stage
    __shared__ float Bs[2 * BUFS];

    const int lane = threadIdx.x & 31;
    const int half = lane >> 4;       // 0 -> K,K+1   1 -> K+2,K+3
    const int l    = lane & 15;

    const int tilesM = (M + 15) >> 4;
    int tm = blockIdx.x * 8 + (threadIdx.x >> 5);   // this wave's M tile
    const bool mvalid = tm < tilesM;
    if (tm >= tilesM) tm = tilesM - 1;              // clamp: stay in barriers
    const int m0 = tm << 4;
    const int n0 = blockIdx.y * BN;

    int am = m0 + l;                  // A row this lane reads
    if (am >= M) am = M - 1;          // clamped, always in-bounds
    const float* __restrict__ aPtr = A + (long)am * K + 2 * half;

    // Async stage of packed-B pair rows [k0/2, k0/2+PR) x [n0, n0+BN) -> buffer
    auto stage = [&](int k0, int buf) {
        const float* bsrc = B + (long)(k0 / 2) * (2 * Nld) + n0 * 2;
        for (int c = threadIdx.x; c < CH; c += 256) {
            const int pr = c / (BN / 2);
            const int q  = c % (BN / 2);
            unsigned lds = (unsigned)(size_t)(&Bs[buf * BUFS + pr * LDP + q * 4]);
            unsigned long long ga =
                (unsigned long long)(size_t)(bsrc + (long)pr * (2 * Nld) + q * 4);
            asm volatile("global_load_async_to_lds_b128 %0, %1, off"
                         :: "v"(lds), "v"(ga) : "memory");
        }
    };

    v8f acc[NT];
#pragma unroll
    for (int nt = 0; nt < NT; ++nt) acc[nt] = (v8f){0.f,0.f,0.f,0.f,0.f,0.f,0.f,0.f};

    const int nChunks = K / KC;
    stage(0, 0);
    for (int c = 0; c < nChunks; ++c) {
        asm volatile("s_wait_asynccnt 0x0" ::: "memory");  // my copies landed
        __syncthreads();                                   // everyone's landed
        if (c + 1 < nChunks) stage((c + 1) * KC, (c + 1) & 1);

        const float* __restrict__ ap = aPtr + c * KC;
        const float* __restrict__ bs = &Bs[(c & 1) * BUFS + (half * LDP) + 2 * l];
#pragma unroll
        for (int kk = 0; kk < KC; kk += 4) {
            const v2f a = *(const v2f*)(ap + kk);
#pragma unroll
            for (int nt = 0; nt < NT; ++nt) {
                const v2f b = *(const v2f*)(bs + (kk >> 1) * LDP + nt * 32);
                acc[nt] = __builtin_amdgcn_wmma_f32_16x16x4_f32(
                    false, a, false, b, (short)0, acc[nt], false, false);
            }
        }
    }

    if (!mvalid) return;              // after last barrier: safe
#pragma unroll
    for (int nt = 0; nt < NT; ++nt) {
        const int bn = n0 + nt * 16 + l;
        if (bn >= Nst) continue;
        const float bv = bias ? bias[bn] : 0.0f;
#pragma unroll
        for (int v = 0; v < 8; ++v) {
            const int m = m0 + v + half * 8;
            if (m >= M) continue;
            float x = acc[nt][v] + bv;
            if (relu) x = fmaxf(x, 0.0f);
            C[(long)m * Nst + bn] = x;
        }
    }
}

// ---------------------------------------------------------------------------
// Edge aggregation: dst[row[e], :] += src[col[e], :] * dis[row]*dis[col]
// One thread per (edge, float4 chunk).  D % 4 == 0 (256 and 40 both qualify).
// ---------------------------------------------------------------------------
__global__ void agg_kernel(const float* __restrict__ src, float* __restrict__ dst,
                           const int* __restrict__ row, const int* __restrict__ col,
                           const float* __restrict__ dis, long total, int dq, int D) {
    long idx = (long)blockIdx.x * blockDim.x + threadIdx.x;
    if (idx >= total) return;
    const long e  = idx / dq;
    const int  c4 = (int)(idx % dq);
    const int  r  = row[e];
    const int  cl = col[e];
    const float norm = dis[r] * dis[cl];
    const float4 v = ((const float4*)(src + (long)cl * D))[c4];
    float* d = dst + (long)r * D + c4 * 4;
    atomicAdd(d + 0, v.x * norm);
    atomicAdd(d + 1, v.y * norm);
    atomicAdd(d + 2, v.z * norm);
    atomicAdd(d + 3, v.w * norm);
}

// dst[i,f] = act(dst[i,f] + src[i,f] * dis[i]^2 + bias[f])
__global__ void self_bias_kernel(float* __restrict__ dst, const float* __restrict__ src,
                                 const float* __restrict__ dis, const float* __restrict__ bias,
                                 long total, int D, int relu) {
    long idx = (long)blockIdx.x * blockDim.x + threadIdx.x;
    if (idx >= total) return;
    const long i = idx / D;
    const int  f = (int)(idx % D);
    const float sc = dis[i] * dis[i];
    float x = dst[idx] + src[idx] * sc + bias[f];
    if (relu) x = fmaxf(x, 0.0f);
    dst[idx] = x;
}

// ---------------------------------------------------------------------------
static inline int grid1d(long total, int block) { return (int)((total + block - 1) / block); }

extern "C" void kernel_launch(void* const* d_in, const int* in_sizes, int n_in,
                              void* d_out, int out_size, void* d_ws, size_t ws_size,
                              hipStream_t stream) {
    // Inputs (setup_inputs order)
    const float* x     = (const float*)d_in[0];   // [N, F]
    const int*   ei    = (const int*)  d_in[1];   // [2, E] (int32 per harness convention)
    const float* W_mlp = (const float*)d_in[2];   // [F, Hm]
    const float* b_mlp = (const float*)d_in[3];   // [Hm]
    const float* W1    = (const float*)d_in[4];   // [Hm, Hg]
    const float* b1    = (const float*)d_in[5];   // [Hg]
    const float* W2    = (const float*)d_in[6];   // [Hg, C]
    const float* b2    = (const float*)d_in[7];   // [C]

    const long E  = in_sizes[1] / 2;
    const int  Hm = in_sizes[3];                  // 256
    const int  F  = in_sizes[2] / Hm;             // 512
    const int  Hg = in_sizes[5];                  // 256
    const int  NC = in_sizes[7];                  // 40
    const int  N  = in_sizes[0] / F;              // 100000
    const int  NCp = ((NC + 47) / 48) * 48;       // pad to 48 (= 16*NT, NT=3)

    const int* row = ei;
    const int* col = ei + E;

    // Workspace layout (floats)
    const size_t nAlign = ((size_t)N + 63) & ~(size_t)63;
    float* dis    = (float*)d_ws;
    float* deg    = dis + nAlign;
    float* bufA   = deg + nAlign;                 // [N, 256]
    float* bufB   = bufA + (size_t)N * Hm;        // [N, 256]
    float* wp_mlp = bufB + (size_t)N * Hm;        // packed [F/2, Hm, 2]
    float* wp_1   = wp_mlp + (size_t)F * Hm;      // packed [Hm/2, Hg, 2]
    float* wp_2   = wp_1 + (size_t)Hm * Hg;       // packed [Hg/2, NCp, 2]

    float* out = (float*)d_out;

    // 1) degrees + symmetric-norm factors; pack all weights (pair-interleave)
    hipMemsetAsync(deg, 0, (size_t)N * sizeof(float), stream);
    deg_kernel<<<grid1d(E, 256), 256, 0, stream>>>(col, deg, E);
    dis_kernel<<<grid1d(N, 256), 256, 0, stream>>>(deg, dis, N);
    pack_pairs_kernel<<<grid1d((long)F * Hm, 256), 256, 0, stream>>>(W_mlp, wp_mlp, F, Hm, Hm);
    pack_pairs_kernel<<<grid1d((long)Hm * Hg, 256), 256, 0, stream>>>(W1, wp_1, Hm, Hg, Hg);
    pack_pairs_kernel<<<grid1d((long)Hg * NCp, 256), 256, 0, stream>>>(W2, wp_2, Hg, NC, NCp);

    const int tilesM  = (N + 15) / 16;
    const int mBlocks = (tilesM + 7) / 8;

    // 2) h0 = relu(X @ W_mlp + b_mlp) -> bufA       [N,512]x[512,256]
    wmma_gemm_f32<4><<<dim3(mBlocks, Hm / 64), 256, 0, stream>>>(
        x, wp_mlp, b_mlp, bufA, N, F, Hm, Hm, /*relu=*/1);

    // 3) t1 = h0 @ W1 -> bufB                       [N,256]x[256,256]
    wmma_gemm_f32<4><<<dim3(mBlocks, Hg / 64), 256, 0, stream>>>(
        bufA, wp_1, nullptr, bufB, N, Hm, Hg, Hg, /*relu=*/0);

    // 4) aggregate t1 over edges into bufA (zeroed), add self term + b1, relu -> h1 in bufA
    hipMemsetAsync(bufA, 0, (size_t)N * Hg * sizeof(float), stream);
    {
        long total = E * (Hg / 4);
        agg_kernel<<<grid1d(total, 256), 256, 0, stream>>>(bufB, bufA, row, col, dis,
                                                           total, Hg / 4, Hg);
        long tot2 = (long)N * Hg;
        self_bias_kernel<<<grid1d(tot2, 256), 256, 0, stream>>>(bufA, bufB, dis, b1,
                                                                tot2, Hg, /*relu=*/1);
    }

    // 5) t2 = h1 @ W2 -> bufB                       [N,256]x[256,40]
    wmma_gemm_f32<3><<<dim3(mBlocks, NCp / 48), 256, 0, stream>>>(
        bufA, wp_2, nullptr, bufB, N, Hg, NCp, NC, /*relu=*/0);

    // 6) aggregate t2 into d_out (zeroed), add self term + b2
    hipMemsetAsync(out, 0, (size_t)out_size * sizeof(float), stream);
    {
        long total = E * (NC / 4);
        agg_kernel<<<grid1d(total, 256), 256, 0, stream>>>(bufB, out, row, col, dis,
                                                           total, NC / 4, NC);
        long tot2 = (long)N * NC;
        self_bias_kernel<<<grid1d(tot2, 256), 256, 0, stream>>>(out, bufB, dis, b2,
                                                                tot2, NC, /*relu=*/0);
    }
}